// ComplexDotProduct_42262478193150
// MI455X (gfx1250) — compile-verified
//
#include <hip/hip_runtime.h>

typedef float v2f __attribute__((ext_vector_type(2)));
typedef float v8f __attribute__((ext_vector_type(8)));

// Problem constants (match reference)
constexpr int Cch = 128;   // channels (K)
constexpr int Nn  = 1024;  // positions (independent GEMMs)
constexpr int Oo  = 512;   // outputs

// Tiling
constexpr int TN = 8;        // n per workgroup (one per wave)
constexpr int TO = 32;       // o per workgroup (2 x 16)
constexpr int KC = 8;        // K chunk staged in LDS
constexpr int KP2 = KC + 1;  // padded k stride (odd -> bank-conflict free)
constexpr int NBLK = Nn / TN;   // 128
constexpr int OBLK = Oo / TO;   // 16
constexpr int NCHUNK = Cch / KC; // 16

// LDS buffer layout (floats), per ping-pong buffer:
//   A comp: [npair(4)][b(64)][k(KP2)][2]  -> n-pair innermost (async b64 lands contiguously)
//   B comp: [npair(4)][o(32)][k(KP2)][2]
constexpr int A_COMP = (TN / 2) * 64 * KP2 * 2;   // 4608
constexpr int B_COMP = (TN / 2) * TO * KP2 * 2;   // 2304
constexpr int OFF_ARE = 0;
constexpr int OFF_AIM = A_COMP;
constexpr int OFF_BRE = 2 * A_COMP;
constexpr int OFF_BIM = 2 * A_COMP + B_COMP;
constexpr int BUF_FLOATS = 2 * A_COMP + 2 * B_COMP;      // 13824
constexpr unsigned LDS_BYTES = 2u * BUF_FLOATS * 4u;     // 110592 (double buffer)

// Async global->LDS copy, 8 bytes per lane (CDNA5 Tensor/async path, ASYNCcnt).
__device__ __forceinline__ void async_b64_rt(unsigned lds_byte, const float* g) {
    asm volatile("global_load_async_to_lds_b64 %0, %1, off"
                 :: "v"(lds_byte), "v"(g) : "memory");
}
// Non-temporal flavor for streamed-once w data (protect x residency in L2).
__device__ __forceinline__ void async_b64_nt(unsigned lds_byte, const float* g) {
    asm volatile("global_load_async_to_lds_b64 %0, %1, off th:TH_LOAD_NT"
                 :: "v"(lds_byte), "v"(g) : "memory");
}

__launch_bounds__(256)
__global__ void ComplexDotProduct_42262478193150_kernel(
    const float* __restrict__ x_re, const float* __restrict__ x_im,
    const float* __restrict__ w_re, const float* __restrict__ w_im,
    const float* __restrict__ b_re, const float* __restrict__ b_im,
    float* __restrict__ out)
{
    extern __shared__ float lds[];   // 2 * BUF_FLOATS floats (double buffer)

    const int t    = threadIdx.x;
    const int wave = t >> 5;          // 0..7 -> owns n = n0 + wave
    const int lane = t & 31;
    const int l15   = lane & 15;
    const int lhalf = lane >> 4;

    const int nblk = blockIdx.x % NBLK;   // n fast in grid -> L2 line sharing
    const int oblk = blockIdx.x / NBLK;
    const int n0 = nblk * TN;
    const int o0 = oblk * TO;

    // LDS byte offset of dynamic-shared base (generic addr low 32 bits = LDS offset)
    const unsigned ldsBase = (unsigned)(uintptr_t)(void*)&lds[0];

    // Accumulators: 8 tiles (4 m-tiles x 2 o-tiles) x {re, im}
    v8f acc_re[8], acc_im[8];
    const v8f vzero = {0.f, 0.f, 0.f, 0.f, 0.f, 0.f, 0.f, 0.f};
#pragma unroll
    for (int i = 0; i < 8; ++i) { acc_re[i] = vzero; acc_im[i] = vzero; }

    // -------- async staging of one K chunk into ping-pong buffer ----------
    auto stage = [&](int kc, int buf) {
        const int c0 = kc * KC;
        const unsigned bufB = ldsBase + (unsigned)buf * (unsigned)(BUF_FLOATS * 4);
        // A (x): 4 npair x 64 b x 8 c float2 per component -> 8 iters of 256 thr
#pragma unroll
        for (int i = 0; i < 8; ++i) {
            int flat  = i * 256 + t;
            int npair = flat & 3;            // fast -> 32B-contiguous global
            int ci    = (flat >> 2) & 7;
            int bb    = flat >> 5;           // 0..63
            size_t g = (size_t)bb * (Cch * Nn) + (size_t)(c0 + ci) * Nn
                     + (size_t)(n0 + npair * 2);
            unsigned lofs = (unsigned)(((npair * 64 + bb) * KP2 + ci) * 2) * 4u;
            async_b64_rt(bufB + OFF_ARE * 4u + lofs, x_re + g);
            async_b64_rt(bufB + OFF_AIM * 4u + lofs, x_im + g);
        }
        // B (w): 4 npair x 32 o x 8 c float2 per component -> 4 iters
#pragma unroll
        for (int i = 0; i < 4; ++i) {
            int flat  = i * 256 + t;
            int npair = flat & 3;
            int ci    = (flat >> 2) & 7;
            int oo    = flat >> 5;           // 0..31
            size_t g = (size_t)(o0 + oo) * (Cch * Nn) + (size_t)(c0 + ci) * Nn
                     + (size_t)(n0 + npair * 2);
            unsigned lofs = (unsigned)(((npair * TO + oo) * KP2 + ci) * 2) * 4u;
            async_b64_nt(bufB + OFF_BRE * 4u + lofs, w_re + g);
            async_b64_nt(bufB + OFF_BIM * 4u + lofs, w_im + g);
        }
        // 24 async b64 issued per thread per chunk
    };

    // -------- WMMA compute on one staged chunk ----------------------------
    const int npw = wave >> 1;   // this wave's n-pair
    const int par = wave & 1;    // even/odd n within the pair
    auto compute = [&](int buf) {
        const float* bp = &lds[buf * BUF_FLOATS];
#pragma unroll
        for (int mk = 0; mk < 2; ++mk) {
            // f32 WMMA fragment: lane holds k = kloc, kloc+1 ; row/col = lane&15
            const int kloc = mk * 4 + 2 * lhalf;
            v2f bfr[2], bfi[2];
#pragma unroll
            for (int ot = 0; ot < 2; ++ot) {
                int ib = ((npw * TO + ot * 16 + l15) * KP2 + kloc) * 2 + par;
                bfr[ot].x = bp[OFF_BRE + ib]; bfr[ot].y = bp[OFF_BRE + ib + 2];
                bfi[ot].x = bp[OFF_BIM + ib]; bfi[ot].y = bp[OFF_BIM + ib + 2];
            }
#pragma unroll
            for (int mt = 0; mt < 4; ++mt) {
                int ia = ((npw * 64 + mt * 16 + l15) * KP2 + kloc) * 2 + par;
                v2f ar, ai, an;
                ar.x = bp[OFF_ARE + ia]; ar.y = bp[OFF_ARE + ia + 2];
                ai.x = bp[OFF_AIM + ia]; ai.y = bp[OFF_AIM + ia + 2];
                an.x = -ai.x; an.y = -ai.y;   // re = rr - ii via negated a_im
#pragma unroll
                for (int ot = 0; ot < 2; ++ot) {
                    const int x = mt * 2 + ot;
                    acc_re[x] = __builtin_amdgcn_wmma_f32_16x16x4_f32(
                        false, ar, false, bfr[ot], (short)0, acc_re[x], false, false);
                    acc_re[x] = __builtin_amdgcn_wmma_f32_16x16x4_f32(
                        false, an, false, bfi[ot], (short)0, acc_re[x], false, false);
                    acc_im[x] = __builtin_amdgcn_wmma_f32_16x16x4_f32(
                        false, ar, false, bfi[ot], (short)0, acc_im[x], false, false);
                    acc_im[x] = __builtin_amdgcn_wmma_f32_16x16x4_f32(
                        false, ai, false, bfr[ot], (short)0, acc_im[x], false, false);
                }
            }
        }
    };

    // -------- software pipeline: prefetch k+1 while computing k -----------
    stage(0, 0);
    for (int kc = 0; kc < NCHUNK; ++kc) {
        const int cur = kc & 1;
        if (kc + 1 < NCHUNK) {
            stage(kc + 1, cur ^ 1);   // overwrites buffer read in previous iter (fenced below)
            // <=24 outstanding == the 24 just issued -> chunk kc fully in LDS
            asm volatile("s_wait_asynccnt 24" ::: "memory");
        } else {
            asm volatile("s_wait_asynccnt 0" ::: "memory");
        }
        __syncthreads();              // publish chunk kc to all waves
        compute(cur);
        __syncthreads();              // all reads of buf 'cur' done before overwrite
    }

    // -------- epilogue: bias add + interleaved complex store --------------
    const int n = n0 + wave;
#pragma unroll
    for (int ot = 0; ot < 2; ++ot) {
        const int og = o0 + ot * 16 + l15;   // lane&15 -> output column o
        const float br = b_re[og];
        const float bi = b_im[og];
#pragma unroll
        for (int mt = 0; mt < 4; ++mt) {
            const int x = mt * 2 + ot;
#pragma unroll
            for (int j = 0; j < 8; ++j) {
                const int bb = mt * 16 + j + 8 * lhalf;  // D row layout
                const size_t oi = ((size_t)bb * Oo + og) * Nn + n;
                v2f o2;
                o2.x = acc_re[x][j] + br;
                o2.y = acc_im[x][j] + bi;
                *(v2f*)(out + oi * 2) = o2;
            }
        }
    }
}

extern "C" void kernel_launch(void* const* d_in, const int* in_sizes, int n_in,
                              void* d_out, int out_size, void* d_ws, size_t ws_size,
                              hipStream_t stream) {
    (void)in_sizes; (void)n_in; (void)out_size; (void)d_ws; (void)ws_size;
    const float* x_re = (const float*)d_in[0];
    const float* x_im = (const float*)d_in[1];
    const float* w_re = (const float*)d_in[2];
    const float* w_im = (const float*)d_in[3];
    const float* b_re = (const float*)d_in[4];
    const float* b_im = (const float*)d_in[5];

    dim3 grid(NBLK * OBLK);   // 2048 workgroups, n-fastest
    dim3 block(256);          // 8 wave32
    ComplexDotProduct_42262478193150_kernel<<<grid, block, LDS_BYTES, stream>>>(
        x_re, x_im, w_re, w_im, b_re, b_im, (float*)d_out);
}